// TritonKroneckerLinear_32847909880414
// MI455X (gfx1250) — compile-verified
//
#include <hip/hip_runtime.h>

// Kronecker linear  Y = X @ kron(A,B)^T + bias  on gfx1250 (MI455X).
//   stage 1: T_m = X_m(64x64) * B^T            (V_WMMA_F32_16X16X4_F32)
//   stage 2: Y_m = A * T_m + bias              (V_WMMA_F32_16X16X4_F32)
// X tiles stream in via the Tensor Data Mover (tensor_load_to_lds) with
// double buffering; T never touches HBM. Right-hand matrices use a K-pair
// interleaved LDS layout so each WMMA B-fragment is one aligned ds_load_b64.

typedef __attribute__((ext_vector_type(2))) float v2f;
typedef __attribute__((ext_vector_type(8))) float v8f;
typedef __attribute__((ext_vector_type(4))) unsigned int v4u;
typedef __attribute__((ext_vector_type(4))) int v4i;
typedef __attribute__((ext_vector_type(8))) int v8i;

#define XLD  68   // leading dim (dwords) of row-major X/A tiles in LDS
#define RPLD 160  // dwords per K-pair row of interleaved R matrices (128+32 pad)

#if __has_builtin(__builtin_amdgcn_tensor_load_to_lds) && \
    __has_builtin(__builtin_amdgcn_s_wait_tensorcnt)
#define USE_TDM 1
#else
#define USE_TDM 0
#endif

// Interleaved R layout: element R[k][n] lives at dword
//   ridx(k>>1, n) + (k&1),  ridx(p,n) = p*RPLD + ((p>>2)&1)*32 + 2n
// -> fragment {R[k][n], R[k+1][n]} is one 8-byte-aligned b64 access.
__device__ __forceinline__ int ridx(int p, int n) {
  return p * RPLD + (((p >> 2) & 1) << 5) + 2 * n;
}

#if USE_TDM
// One-wave TDM: DMA a contiguous 64x64 f32 tile (16 KB) global -> LDS,
// hardware-padding each 64-dword row by 4 dwords (XLD = 68).
__device__ __forceinline__ void tdm_load_x(const float* gsrc, unsigned lds_byte) {
  unsigned long long ga = (unsigned long long)(uintptr_t)gsrc;
  v4u g0;
  g0.x = 1u;                                            // count=1, user mode
  g0.y = lds_byte;                                      // lds_addr
  g0.z = (unsigned)ga;                                  // global_addr[31:0]
  g0.w = (unsigned)((ga >> 32) & 0x01FFFFFFu) | (2u << 30); // ga[56:32] | type=2
  v8i g1;
  g1[0] = (2 << 16)    // data_size = 4 bytes
        | (1 << 20)    // pad_enable
        | (5 << 22)    // pad_interval: 64 dwords
        | (3 << 25);   // pad_amount:   4 dwords
  g1[1] = (64 << 16);  // tensor_dim0 = 64 (low 16 into [31:16])
  g1[2] = (64 << 16);  // tensor_dim0 hi = 0 | tensor_dim1 = 64
  g1[3] = (64 << 16);  // tensor_dim1 hi = 0 | tile_dim0 = 64
  g1[4] = 64;          // tile_dim1 = 64, tile_dim2 = 0
  g1[5] = 64;          // tensor_dim0_stride = 64
  g1[6] = 0;
  g1[7] = 0;
  v4i gz4 = {0, 0, 0, 0};                 // groups 2/3 unused (2D tensor)
  v8i gz8 = {0, 0, 0, 0, 0, 0, 0, 0};     // extra group (6-arg toolchain form)
  __builtin_amdgcn_tensor_load_to_lds(g0, g1, gz4, gz4, gz8, 0);
}
#endif

// Two 16x16 tiles (tM,tN),(tM,tN+16) of C = L(64x64) * R(64x64).
// L row-major (leading dim XLD); R in interleaved layout.
__device__ __forceinline__ void gemm64(const float* __restrict__ Lp,
                                       const float* __restrict__ Rp,
                                       int tM, int tN, int half, int l16,
                                       v8f& c0, v8f& c1) {
#pragma unroll
  for (int k0 = 0; k0 < 64; k0 += 4) {
    // A-frag: row tM+l16, K = k0 + 2*half + {0,1}
    v2f a = *(const v2f*)(Lp + (tM + l16) * XLD + k0 + 2 * half);
    // B-frags: cols tN[+16]+l16, same K pair -> single b64 each
    const float* bp = Rp + ridx((k0 >> 1) + half, tN + l16);
    v2f b0 = *(const v2f*)bp;
    v2f b1 = *(const v2f*)(bp + 32);
    c0 = __builtin_amdgcn_wmma_f32_16x16x4_f32(false, a, false, b0,
                                               (short)0, c0, false, false);
    c1 = __builtin_amdgcn_wmma_f32_16x16x4_f32(false, a, false, b1,
                                               (short)0, c1, false, false);
  }
}

__global__ __launch_bounds__(256, 1) void kron_wmma_f32_kernel(
    const float* __restrict__ x,     // [8192, 4096] = [m, ka*64+kb]
    const float* __restrict__ A,     // [64, 64]     = [na, ka]
    const float* __restrict__ B,     // [64, 64]     = [nb, kb]
    const float* __restrict__ bias,  // [4096]       = [na*64+nb]
    float* __restrict__ out)         // [8192, 4096] = [m, na*64+nb]
{
  __shared__ float ldsX[2][64 * XLD];   // double-buffered X_m (rows = ka)
  __shared__ float ldsA[64 * XLD];      // A row-major
  __shared__ float ldsBtS[32 * RPLD];   // B^T, interleaved: [kb-pair][nb]
  __shared__ float ldsTS[32 * RPLD];    // T,   interleaved: [ka-pair][nb]

  const int t = threadIdx.x;

  // Stage A and B^T once per workgroup.
#pragma unroll
  for (int i = 0; i < 16; ++i) {
    int flat = t + 256 * i;          // A: na*64+ka   B: nb*64+kb
    int r = flat >> 6;
    int c = flat & 63;
    ldsA[r * XLD + c] = A[flat];
    ldsBtS[ridx(c >> 1, r) + (c & 1)] = B[flat];   // Bt[kb][nb] = B[nb][kb]
  }

  const int wave = t >> 5;
  const int lane = t & 31;
  const int half = lane >> 4;
  const int l16  = lane & 15;
  const int tM   = (wave >> 1) * 16;   // 4 tile-rows, 2 waves per row
  const int tN   = (wave & 1) * 32;    // 2 adjacent col-tiles per wave

  // Bias is fixed per lane across all m: preload once.
  float bs0[8], bs1[8];
#pragma unroll
  for (int j = 0; j < 8; ++j) {
    int row = tM + 8 * half + j;       // C layout: VGPR j -> M = j + 8*half
    bs0[j] = bias[row * 64 + tN + l16];
    bs1[j] = bias[row * 64 + tN + 16 + l16];
  }

  const int stride = (int)gridDim.x;
  int m = (int)blockIdx.x;

#if USE_TDM
  if (m < 8192 && wave == 0)
    tdm_load_x(x + (size_t)m * 4096, (unsigned)(uintptr_t)&ldsX[0][0]);
#endif

  int it = 0;
  for (; m < 8192; m += stride, ++it) {
    const int cur = it & 1;

#if USE_TDM
    // Prefetch next X tile into the other buffer, then wait for current.
    if (wave == 0) {
      if (m + stride < 8192) {
        tdm_load_x(x + (size_t)(m + stride) * 4096,
                   (unsigned)(uintptr_t)&ldsX[cur ^ 1][0]);
        __builtin_amdgcn_s_wait_tensorcnt(1);
      } else {
        __builtin_amdgcn_s_wait_tensorcnt(0);
      }
    }
    __syncthreads();   // X[cur] visible; also guards ldsTS/X-buffer reuse
#else
    __syncthreads();
    const float* xm = x + (size_t)m * 4096;
#pragma unroll
    for (int i = 0; i < 4; ++i) {
      int f4 = t + 256 * i;
      int e  = f4 << 2;
      *(float4*)&ldsX[cur][(e >> 6) * XLD + (e & 63)] = ((const float4*)xm)[f4];
    }
    __syncthreads();
#endif

    // Stage 1: T = X_m * B^T
    v8f c0 = {}, c1 = {};
    gemm64(&ldsX[cur][0], ldsBtS, tM, tN, half, l16, c0, c1);
#pragma unroll
    for (int j = 0; j < 8; j += 2) {   // rows j, j+1 share a K-pair -> b64
      int p = (tM + 8 * half + j) >> 1;
      v2f t0; t0.x = c0[j]; t0.y = c0[j + 1];
      v2f t1; t1.x = c1[j]; t1.y = c1[j + 1];
      *(v2f*)(ldsTS + ridx(p, tN + l16))      = t0;
      *(v2f*)(ldsTS + ridx(p, tN + 16 + l16)) = t1;
    }
    __syncthreads();

    // Stage 2: Y_m = A * T + bias
    v8f d0 = {}, d1 = {};
    gemm64(ldsA, ldsTS, tM, tN, half, l16, d0, d1);

    float* om = out + (size_t)m * 4096;
#pragma unroll
    for (int j = 0; j < 8; ++j) {
      int row = tM + 8 * half + j;
      om[row * 64 + tN + l16]      = d0[j] + bs0[j];
      om[row * 64 + tN + 16 + l16] = d1[j] + bs1[j];
    }
  }
}

extern "C" void kernel_launch(void* const* d_in, const int* in_sizes, int n_in,
                              void* d_out, int out_size, void* d_ws, size_t ws_size,
                              hipStream_t stream) {
  (void)in_sizes; (void)n_in; (void)out_size; (void)d_ws; (void)ws_size;
  const float* x    = (const float*)d_in[0];
  const float* A    = (const float*)d_in[1];
  const float* B    = (const float*)d_in[2];
  const float* bias = (const float*)d_in[3];
  float* out        = (float*)d_out;

  dim3 grid(2048);   // 8192 rows / 2048 WGs = 4 m-iterations per WG
  dim3 block(256);   // 8 wave32s
  hipLaunchKernelGGL(kron_wmma_f32_kernel, grid, block, 0, stream,
                     x, A, B, bias, out);
}